// GINLayer_43765716746314
// MI455X (gfx1250) — compile-verified
//
#include <hip/hip_runtime.h>

#define N_NODES 50000
#define N_EDGES 600000
#define DIM     128
#define DIM2    256
#define BN_EPS  1e-5f

typedef __attribute__((ext_vector_type(2))) float v2f;
typedef __attribute__((ext_vector_type(8))) float v8f;

// ---------------------------------------------------------------------------
// Kernel 1: zero agg accumulator + BN column sum/sumsq accumulators
// ---------------------------------------------------------------------------
__global__ void zero_kernel(float* __restrict__ agg, float* __restrict__ sums,
                            int total) {
  int idx = blockIdx.x * blockDim.x + threadIdx.x;
  if (idx < total) agg[idx] = 0.0f;
  if (idx < 2 * DIM) sums[idx] = 0.0f;
}

// ---------------------------------------------------------------------------
// Kernel 2: per-edge message build + scatter-add.
// One wave32 per edge; each lane owns 4 consecutive features (float4).
// E = 600000 = 75000 blocks * 8 waves exactly -> no tail, EXEC stays full.
// ---------------------------------------------------------------------------
__global__ void edge_scatter(const float* __restrict__ nf,
                             const float* __restrict__ emb0,
                             const float* __restrict__ emb1,
                             const int* __restrict__ src,
                             const int* __restrict__ dst,
                             const int* __restrict__ ef0,
                             const int* __restrict__ ef1,
                             float* __restrict__ agg) {
  int e    = blockIdx.x * 8 + (threadIdx.x >> 5);
  int lane = threadIdx.x & 31;
  int s  = src[e];
  int d  = dst[e];
  int f0 = ef0[e];
  int f1 = ef1[e];

  const float4 a = *((const float4*)(nf   + (size_t)s  * DIM) + lane);
  const float4 b = *((const float4*)(emb0 + (size_t)f0 * DIM) + lane);
  const float4 c = *((const float4*)(emb1 + (size_t)f1 * DIM) + lane);

  float m0 = a.x + b.x + c.x;
  float m1 = a.y + b.y + c.y;
  float m2 = a.z + b.z + c.z;
  float m3 = a.w + b.w + c.w;

  float* p = agg + (size_t)d * DIM + lane * 4;
  atomicAdd(p + 0, m0);
  atomicAdd(p + 1, m1);
  atomicAdd(p + 2, m2);
  atomicAdd(p + 3, m3);
}

// ---------------------------------------------------------------------------
// Kernel 3: h1 = relu(agg @ W1 + b1)   [N,128] x [128,256] -> [N,256]
// fp32 WMMA 16x16x4. One wave computes a 16x64 output strip (4 tiles),
// reusing the A operand across the 4 WMMAs per k-step.
// A layout (16x4 f32): lanes 0-15 = M, {VGPR0,VGPR1} = {K+2h, K+1+2h}, h=lane>>4
// B layout (4x16 f32): lanes 0-15 = N, {VGPR0,VGPR1} = rows {K+2h, K+1+2h}
// C/D layout (16x16 f32): VGPR j: lanes 0-15 -> M=j, lanes 16-31 -> M=j+8
// ---------------------------------------------------------------------------
__global__ void gemm1_relu(const float* __restrict__ agg,
                           const float* __restrict__ W1,
                           const float* __restrict__ b1,
                           float* __restrict__ h1) {
  const int ROW_TILES  = N_NODES / 16;   // 3125
  const int COL_GROUPS = DIM2 / 64;      // 4
  int wid = blockIdx.x * (blockDim.x >> 5) + (threadIdx.x >> 5);
  if (wid >= ROW_TILES * COL_GROUPS) return;   // wave-uniform: EXEC all-1s

  int rowTile = wid / COL_GROUPS;
  int cg      = wid % COL_GROUPS;
  int lane    = threadIdx.x & 31;
  int half    = lane >> 4;
  int l       = lane & 15;

  const float* Arow = agg + (size_t)(rowTile * 16 + l) * DIM;
  int n0 = cg * 64 + l;

  v8f acc[4] = {};
  for (int k = 0; k < DIM; k += 4) {
    v2f a = *(const v2f*)(Arow + k + 2 * half);      // {K+2h, K+1+2h} of row M=l
    const float* B0 = W1 + (size_t)(k + 2 * half) * DIM2;
    const float* B1 = B0 + DIM2;
#pragma unroll
    for (int t = 0; t < 4; ++t) {
      v2f b;
      b.x = B0[n0 + t * 16];
      b.y = B1[n0 + t * 16];
      acc[t] = __builtin_amdgcn_wmma_f32_16x16x4_f32(
          false, a, false, b, (short)0, acc[t], false, false);
    }
  }

#pragma unroll
  for (int t = 0; t < 4; ++t) {
    int n = cg * 64 + t * 16 + l;
    float bias = b1[n];
#pragma unroll
    for (int j = 0; j < 8; ++j) {
      int m = rowTile * 16 + half * 8 + j;
      float v = acc[t][j] + bias;
      h1[(size_t)m * DIM2 + n] = v > 0.0f ? v : 0.0f;
    }
  }
}

// ---------------------------------------------------------------------------
// Kernel 4: h = h1 @ W2 + b2   [N,256] x [256,128] -> [N,128] (written to out)
// plus per-column sum / sumsq accumulation for BatchNorm.
// ---------------------------------------------------------------------------
__global__ void gemm2_bnstats(const float* __restrict__ h1,
                              const float* __restrict__ W2,
                              const float* __restrict__ b2,
                              float* __restrict__ out,
                              float* __restrict__ colsum,
                              float* __restrict__ colsumsq) {
  const int ROW_TILES  = N_NODES / 16;   // 3125
  const int COL_GROUPS = DIM / 64;       // 2
  int wid = blockIdx.x * (blockDim.x >> 5) + (threadIdx.x >> 5);
  if (wid >= ROW_TILES * COL_GROUPS) return;   // wave-uniform

  int rowTile = wid / COL_GROUPS;
  int cg      = wid % COL_GROUPS;
  int lane    = threadIdx.x & 31;
  int half    = lane >> 4;
  int l       = lane & 15;

  const float* Arow = h1 + (size_t)(rowTile * 16 + l) * DIM2;
  int n0 = cg * 64 + l;

  v8f acc[4] = {};
  for (int k = 0; k < DIM2; k += 4) {
    v2f a = *(const v2f*)(Arow + k + 2 * half);
    const float* B0 = W2 + (size_t)(k + 2 * half) * DIM;
    const float* B1 = B0 + DIM;
#pragma unroll
    for (int t = 0; t < 4; ++t) {
      v2f b;
      b.x = B0[n0 + t * 16];
      b.y = B1[n0 + t * 16];
      acc[t] = __builtin_amdgcn_wmma_f32_16x16x4_f32(
          false, a, false, b, (short)0, acc[t], false, false);
    }
  }

#pragma unroll
  for (int t = 0; t < 4; ++t) {
    int n = cg * 64 + t * 16 + l;
    float bias = b2[n];
    float ls = 0.0f, lss = 0.0f;
#pragma unroll
    for (int j = 0; j < 8; ++j) {
      int m = rowTile * 16 + half * 8 + j;
      float v = acc[t][j] + bias;
      out[(size_t)m * DIM + n] = v;
      ls  += v;
      lss += v * v;
    }
    atomicAdd(&colsum[n], ls);
    atomicAdd(&colsumsq[n], lss);
  }
}

// ---------------------------------------------------------------------------
// Kernel 5: BatchNorm finalize in-place on out.
// ---------------------------------------------------------------------------
__global__ void bn_finalize(float* __restrict__ out,
                            const float* __restrict__ colsum,
                            const float* __restrict__ colsumsq,
                            const float* __restrict__ gamma,
                            const float* __restrict__ beta) {
  int idx = blockIdx.x * blockDim.x + threadIdx.x;
  if (idx >= N_NODES * DIM) return;
  int c = idx & (DIM - 1);
  const float invN = 1.0f / (float)N_NODES;
  float mean = colsum[c] * invN;
  float var  = colsumsq[c] * invN - mean * mean;
  float inv  = rsqrtf(var + BN_EPS);
  out[idx] = (out[idx] - mean) * inv * gamma[c] + beta[c];
}

// ---------------------------------------------------------------------------
extern "C" void kernel_launch(void* const* d_in, const int* in_sizes, int n_in,
                              void* d_out, int out_size, void* d_ws, size_t ws_size,
                              hipStream_t stream) {
  const float* node_feats = (const float*)d_in[0];
  const float* emb0       = (const float*)d_in[1];
  const float* emb1       = (const float*)d_in[2];
  const float* W1         = (const float*)d_in[3];
  const float* b1         = (const float*)d_in[4];
  const float* W2         = (const float*)d_in[5];
  const float* b2         = (const float*)d_in[6];
  const float* gamma      = (const float*)d_in[7];
  const float* beta       = (const float*)d_in[8];
  const int*   src        = (const int*)d_in[9];
  const int*   dst        = (const int*)d_in[10];
  const int*   ef0        = (const int*)d_in[11];
  const int*   ef1        = (const int*)d_in[12];
  float*       out        = (float*)d_out;

  float* ws       = (float*)d_ws;
  float* agg      = ws;                                 // N*128 floats (25.6 MB)
  float* h1       = agg + (size_t)N_NODES * DIM;        // N*256 floats (51.2 MB)
  float* colsum   = h1 + (size_t)N_NODES * DIM2;        // 128 floats
  float* colsumsq = colsum + DIM;                       // 128 floats

  // 1) zero accumulators
  {
    int total = N_NODES * DIM;                          // 6,400,000
    int blocks = (total + 255) / 256;
    zero_kernel<<<blocks, 256, 0, stream>>>(agg, colsum, total);
  }

  // 2) edge gather + scatter-add (one wave per edge, 8 waves/block)
  {
    int blocks = N_EDGES / 8;                           // 75,000 exactly
    edge_scatter<<<blocks, 256, 0, stream>>>(node_feats, emb0, emb1,
                                             src, dst, ef0, ef1, agg);
  }

  // 3) GEMM1 + ReLU (WMMA f32): 3125 row tiles * 4 col groups = 12500 waves
  {
    int waves  = (N_NODES / 16) * (DIM2 / 64);
    int blocks = (waves + 7) / 8;
    gemm1_relu<<<blocks, 256, 0, stream>>>(agg, W1, b1, h1);
  }

  // 4) GEMM2 + BN stats (WMMA f32): 3125 * 2 = 6250 waves
  {
    int waves  = (N_NODES / 16) * (DIM / 64);
    int blocks = (waves + 7) / 8;
    gemm2_bnstats<<<blocks, 256, 0, stream>>>(h1, W2, b2, out, colsum, colsumsq);
  }

  // 5) BatchNorm finalize in-place on out
  {
    int total = N_NODES * DIM;
    int blocks = (total + 255) / 256;
    bn_finalize<<<blocks, 256, 0, stream>>>(out, colsum, colsumsq, gamma, beta);
  }
}